// RGCN_8701603742301
// MI455X (gfx1250) — compile-verified
//
#include <hip/hip_runtime.h>

// R-GCN forward (2 layers, basis decomposition) for MI455X / gfx1250.
// Dense math (basis composition GEMMs) runs on V_WMMA_F32_16X16X4_F32 with
// the small A operand staged in LDS (zero-padded) to keep the WMMA inner loop
// free of exec-mask divergence; graph message passing is fused per-edge
// scatter through L2 atomics.

typedef __attribute__((ext_vector_type(2))) float v2f;
typedef __attribute__((ext_vector_type(8))) float v8f;

static inline int idiv_up(long long a, long long b) { return (int)((a + b - 1) / b); }

// ---------------------------------------------------------------------------
// Utility fills
// ---------------------------------------------------------------------------
__global__ void fill_zero_kernel(float* __restrict__ ptr, long long count) {
  long long i = (long long)blockIdx.x * blockDim.x + threadIdx.x;
  if (i < count) ptr[i] = 0.0f;
}

__global__ void init_out_kernel(float* __restrict__ out,
                                const float* __restrict__ bias2,
                                long long count, int C) {
  long long i = (long long)blockIdx.x * blockDim.x + threadIdx.x;
  if (i < count) out[i] = bias2[i % C];
}

// ---------------------------------------------------------------------------
// Degree (segment counts) over the enriched edge list:
//   originals: seg = p*N + s ; inverses: seg = (p+R)*N + o ; self-loops: deg=1
// ---------------------------------------------------------------------------
__global__ void degree_kernel(const int* __restrict__ s, const int* __restrict__ p,
                              const int* __restrict__ o, float* __restrict__ deg,
                              int E, int N, int R) {
  long long i = (long long)blockIdx.x * blockDim.x + threadIdx.x;
  if (i < E) {
    int si = s[i], pi = p[i], oi = o[i];
    atomicAdd(&deg[(size_t)pi * N + si], 1.0f);
    atomicAdd(&deg[(size_t)(pi + R) * N + oi], 1.0f);
  } else if (i < (long long)E + N) {
    int node = (int)(i - E);
    deg[(size_t)(2 * R) * N + node] = 1.0f;  // each self-loop segment has exactly 1 edge
  }
}

// ---------------------------------------------------------------------------
// C[M x J] = A[M x K] * B[K x J]   (f32, WMMA 16x16x4)
//   M <= 48 (rows zero-padded in LDS), K % 4 == 0 (K <= 64), J % 16 == 0.
// One wave computes a 48x16 C tile: 3 stacked 16x16 accumulators share each
// B fragment so B (the big operand) is streamed exactly once.
// A fragment layout (ISA 7.12.2, 32-bit A 16x4): lane = 16*(k>=2) + m, vgpr = k&1.
// B/C/D: row striped across lanes within a VGPR.
// ---------------------------------------------------------------------------
__global__ void gemm48_wmma_kernel(const float* __restrict__ A,
                                   const float* __restrict__ B,
                                   float* __restrict__ C,
                                   int M, int K, long long J) {
  __shared__ float sA[48 * 64];  // zero-padded A tile, K <= 64

  // Cooperative, zero-padded staging of A into LDS (divergence harmless here).
  const int total = 48 * K;
  for (int idx = threadIdx.x; idx < total; idx += blockDim.x) {
    const int row = idx / K;
    const int kk  = idx - row * K;
    sA[idx] = (row < M) ? A[row * K + kk] : 0.0f;
  }
  __syncthreads();

  const int wave = threadIdx.x >> 5;
  const int lane = threadIdx.x & 31;
  const long long jbase =
      ((long long)blockIdx.x * (blockDim.x >> 5) + wave) * 16;
  if (jbase >= J) return;                 // wave-uniform guard: EXEC all-1s at WMMA
  const int col   = lane & 15;
  const int khalf = (lane >> 4) * 2;      // lanes 0-15 -> k {0,1}; lanes 16-31 -> k {2,3}

  v8f acc[3];
  acc[0] = (v8f){};
  acc[1] = (v8f){};
  acc[2] = (v8f){};

  const float* __restrict__ Bp = B + jbase + col;
  for (int kc = 0; kc < K; kc += 4) {
    const int k0 = kc + khalf;
    v2f b;
    b.x = Bp[(long long)k0 * J];
    b.y = Bp[(long long)(k0 + 1) * J];
#pragma unroll
    for (int mt = 0; mt < 3; ++mt) {
      const int m = mt * 16 + col;
      // 8B-aligned LDS read (K and k0 even): ds_load_b64, no exec divergence.
      const v2f a = *(const v2f*)&sA[m * K + k0];
      acc[mt] = __builtin_amdgcn_wmma_f32_16x16x4_f32(
          false, a, false, b, (short)0, acc[mt], false, false);
    }
  }

  const int rhalf = (lane >> 4) * 8;      // lanes 16-31 hold rows m+8
#pragma unroll
  for (int mt = 0; mt < 3; ++mt) {
#pragma unroll
    for (int v = 0; v < 8; ++v) {
      const int row = mt * 16 + rhalf + v;
      if (row < M) C[(long long)row * J + jbase + col] = acc[mt][v];
    }
  }
}

// ---------------------------------------------------------------------------
// Layer 1 scatter: h[s,:] += val * W1[p, o, :]   (thread per (edge, e))
// ---------------------------------------------------------------------------
__global__ void layer1_scatter_kernel(const int* __restrict__ s, const int* __restrict__ p,
                                      const int* __restrict__ o,
                                      const float* __restrict__ deg,
                                      const float* __restrict__ W1,
                                      float* __restrict__ h,
                                      int E, int N, int R, long long NE) {
  long long gid = (long long)blockIdx.x * blockDim.x + threadIdx.x;
  const long long total = ((long long)2 * E + N) * 16;
  if (gid >= total) return;
  const int e = (int)(gid & 15);
  const long long i = gid >> 4;
  int si, pi, oi;
  if (i < E)              { si = s[i];      pi = p[i];          oi = o[i]; }
  else if (i < 2LL * E)   { long long j = i - E; si = o[j]; pi = p[j] + R; oi = s[j]; }
  else                    { int nd = (int)(i - 2LL * E); si = nd; pi = 2 * R; oi = nd; }
  const float val = 1.0f / deg[(size_t)pi * N + si];
  const float w   = W1[(size_t)pi * NE + (size_t)oi * 16 + e];
  atomicAdd(&h[(size_t)si * 16 + e], val * w);
}

__global__ void relu_bias_kernel(float* __restrict__ h, const float* __restrict__ bias1,
                                 long long count) {
  long long i = (long long)blockIdx.x * blockDim.x + threadIdx.x;
  if (i < count) h[i] = fmaxf(h[i] + bias1[i & 15], 0.0f);
}

// ---------------------------------------------------------------------------
// Layer 2, fully fused:  out[s,c] += val * sum_e h[o,e] * W2[p,e,c]
// (skips materializing m[r_tot, n, e] entirely).  Thread per (edge, c).
// ---------------------------------------------------------------------------
__global__ void layer2_scatter_kernel(const int* __restrict__ s, const int* __restrict__ p,
                                      const int* __restrict__ o,
                                      const float* __restrict__ deg,
                                      const float* __restrict__ h,
                                      const float* __restrict__ W2,  // [r_tot][16*8]
                                      float* __restrict__ out,
                                      int E, int N, int R) {
  long long gid = (long long)blockIdx.x * blockDim.x + threadIdx.x;
  const long long total = ((long long)2 * E + N) * 8;
  if (gid >= total) return;
  const int c = (int)(gid & 7);
  const long long i = gid >> 3;
  int si, pi, oi;
  if (i < E)              { si = s[i];      pi = p[i];          oi = o[i]; }
  else if (i < 2LL * E)   { long long j = i - E; si = o[j]; pi = p[j] + R; oi = s[j]; }
  else                    { int nd = (int)(i - 2LL * E); si = nd; pi = 2 * R; oi = nd; }
  const float val = 1.0f / deg[(size_t)pi * N + si];
  const float* hv = &h[(size_t)oi * 16];
  const float* w  = &W2[(size_t)pi * 128 + c];
  float sum = 0.0f;
#pragma unroll
  for (int e = 0; e < 16; ++e) sum += hv[e] * w[e * 8];
  atomicAdd(&out[(size_t)si * 8 + c], val * sum);
}

// ---------------------------------------------------------------------------
// Host launcher
// ---------------------------------------------------------------------------
extern "C" void kernel_launch(void* const* d_in, const int* in_sizes, int n_in,
                              void* d_out, int out_size, void* d_ws, size_t ws_size,
                              hipStream_t stream) {
  const float* comps1 = (const float*)d_in[0];
  const float* bases1 = (const float*)d_in[1];   // [BASES, N, 16]  -> flat [BASES, N*16]
  const float* comps2 = (const float*)d_in[2];
  const float* bases2 = (const float*)d_in[3];   // [BASES, 16, 8]  -> flat [BASES, 128]
  const float* bias1  = (const float*)d_in[4];
  const float* bias2  = (const float*)d_in[5];
  const int*   s      = (const int*)d_in[6];
  const int*   p      = (const int*)d_in[7];
  const int*   o      = (const int*)d_in[8];

  const int EMB    = in_sizes[4];                        // 16
  const int NUMCLS = in_sizes[5];                        // 8
  const int E      = in_sizes[6];                        // 500000
  const int N      = out_size / NUMCLS;                  // 100000
  const int BASES  = (int)((long long)in_sizes[1] / ((long long)N * EMB)); // 40
  const int r_tot  = in_sizes[0] / BASES;                // 41
  const int R      = (r_tot - 1) / 2;                    // 20

  // Workspace layout (floats, 256B aligned slabs)
  float* ws = (float*)d_ws;
  size_t off = 0;
  auto alloc = [&](size_t nfloats) {
    size_t cur = off;
    off += (nfloats + 63) & ~(size_t)63;
    return cur;
  };
  float* deg = ws + alloc((size_t)r_tot * N);                   // 16.4 MB
  float* h   = ws + alloc((size_t)N * EMB);                     //  6.4 MB
  float* W2  = ws + alloc((size_t)r_tot * EMB * NUMCLS);        //  21 KB
  float* W1  = ws + alloc((size_t)r_tot * N * EMB);             // 262 MB
  (void)ws_size; (void)n_in;

  const long long NE   = (long long)N * EMB;       // 1.6M columns for W1 GEMM
  const long long J2   = (long long)EMB * NUMCLS;  // 128 columns for W2 GEMM
  const long long Eall = (long long)2 * E + N;
  const int B = 256;

  // 0) init accumulators
  fill_zero_kernel<<<idiv_up((long long)r_tot * N, B), B, 0, stream>>>(deg, (long long)r_tot * N);
  fill_zero_kernel<<<idiv_up(NE, B), B, 0, stream>>>(h, NE);

  // 1) degrees of the row-stacked adjacency
  degree_kernel<<<idiv_up((long long)E + N, B), B, 0, stream>>>(s, p, o, deg, E, N, R);

  // 2) basis composition via WMMA: W1 = comps1 x bases1, W2 = comps2 x bases2
  gemm48_wmma_kernel<<<idiv_up(NE, 16LL * (B / 32)), B, 0, stream>>>(
      comps1, bases1, W1, r_tot, BASES, NE);
  gemm48_wmma_kernel<<<idiv_up(J2, 16LL * (B / 32)), B, 0, stream>>>(
      comps2, bases2, W2, r_tot, BASES, J2);

  // 3) layer 1: normalized message scatter + ReLU(+bias)
  layer1_scatter_kernel<<<idiv_up(Eall * 16, B), B, 0, stream>>>(
      s, p, o, deg, W1, h, E, N, R, NE);
  relu_bias_kernel<<<idiv_up(NE, B), B, 0, stream>>>(h, bias1, NE);

  // 4) layer 2 fused: out = bias2 + sum_edges val * (h[o] . W2[p])
  init_out_kernel<<<idiv_up((long long)N * NUMCLS, B), B, 0, stream>>>(
      (float*)d_out, bias2, (long long)N * NUMCLS, NUMCLS);
  layer2_scatter_kernel<<<idiv_up(Eall * 8, B), B, 0, stream>>>(
      s, p, o, deg, h, W2, (float*)d_out, E, N, R);
}